// Length_Regulator_6957847019835
// MI455X (gfx1250) — compile-verified
//
#include <hip/hip_runtime.h>
#include <stdint.h>

// Problem constants (from reference): text_memory (32, 512, 384) f32,
// EXPAND_FACTOR=6, mel_len=3500 -> out (32, 3500, 384) f32.
constexpr int B_    = 32;
constexpr int T_IN  = 512;
constexpr int D_    = 384;           // 1536 bytes per row
constexpr int E_    = 6;
constexpr int T_REP = T_IN * E_;     // 3072
constexpr int T_OUT = 3500;
constexpr int ROWS  = B_ * T_IN;     // 16384 input rows
constexpr int F4_PER_ROW = D_ / 4;   // 96 float4 per input row

// Raw clang vector types (HIP_vector_type wrappers are rejected by builtins).
typedef int   v4i __attribute__((vector_size(16)));
typedef float f4v __attribute__((ext_vector_type(4)));

// Address-space-qualified 128-bit vector pointers matching the builtin
// signature revealed by the round-1 diagnostic:
//   param0: v4i __device__* (addrspace(1)), param1: presumably __shared__ (AS3)
typedef __attribute__((address_space(1))) v4i gv4i_t;
typedef __attribute__((address_space(3))) v4i lv4i_t;

#if __has_builtin(__builtin_amdgcn_global_load_async_to_lds_b128) && \
    __has_builtin(__builtin_amdgcn_global_store_async_from_lds_b128)
#define HAVE_ASYNC_LDS 1
#else
#define HAVE_ASYNC_LDS 0
#endif

__device__ __forceinline__ void wait_async0() {
#if __has_builtin(__builtin_amdgcn_s_wait_asynccnt)
  __builtin_amdgcn_s_wait_asynccnt(0);
#else
  asm volatile("s_wait_asynccnt 0" ::: "memory");
#endif
}

#if HAVE_ASYNC_LDS
__device__ __forceinline__ void async_load16(const float* g, float* l) {
  __builtin_amdgcn_global_load_async_to_lds_b128(
      (gv4i_t*)g, (lv4i_t*)l, /*offset=*/0, /*cpol=*/0);
}
__device__ __forceinline__ void async_store16(float* g, const float* l) {
  __builtin_amdgcn_global_store_async_from_lds_b128(
      (gv4i_t*)g, (lv4i_t*)l, /*offset=*/0, /*cpol=*/0);
}
#endif

// Expand kernel: one wave32 per input row.
// Row = 384 f32 = 1536B = 3 chunks of 512B (32 lanes x 16B).
// The 6 output replicas are contiguous: a flat 9216B = 18 chunks of 512B.
__global__ void __launch_bounds__(256)
length_regulator_expand(const float* __restrict__ in, float* __restrict__ out) {
  const int lane   = threadIdx.x & 31;
  const int wave   = (blockIdx.x * blockDim.x + threadIdx.x) >> 5;
  const int nwaves = (gridDim.x * blockDim.x) >> 5;

#if HAVE_ASYNC_LDS
  __shared__ float smem[8 * D_];                 // 8 waves/block * 1536B = 12KB
  float* lbuf = smem + (int)(threadIdx.x >> 5) * D_;

  for (int r = wave; r < ROWS; r += nwaves) {
    const int b = r >> 9;                        // r / 512
    const int t = r & (T_IN - 1);                // r % 512
    const float* src = in + (size_t)r * D_;
    float* dst = out + ((size_t)b * T_OUT + (size_t)t * E_) * D_;

    // Make sure previous iteration's async stores finished reading LDS.
    wait_async0();

    #pragma unroll
    for (int c = 0; c < 3; ++c)                  // 3 x 512B : row -> LDS
      async_load16(src + c * 128 + lane * 4, lbuf + c * 128 + lane * 4);

    wait_async0();                               // LDS now holds the row

    #pragma unroll
    for (int k = 0; k < 18; ++k)                 // 18 x 512B : LDS -> 6 out rows
      async_store16(dst + k * 128 + lane * 4,
                    lbuf + (k % 3) * 128 + lane * 4);
  }
  wait_async0();                                 // drain before wave exit
#else
  // Fallback: plain float4 register copy (same HBM traffic, VGPR round-trip).
  const float4* in4 = reinterpret_cast<const float4*>(in);
  float4* out4 = reinterpret_cast<float4*>(out);
  for (int r = wave; r < ROWS; r += nwaves) {
    const int b = r >> 9;
    const int t = r & (T_IN - 1);
    const float4* src = in4 + (size_t)r * F4_PER_ROW;
    float4* dst = out4 + ((size_t)b * T_OUT + (size_t)t * E_) * F4_PER_ROW;
    const float4 v0 = src[lane];
    const float4 v1 = src[lane + 32];
    const float4 v2 = src[lane + 64];
    #pragma unroll
    for (int rep = 0; rep < E_; ++rep) {
      float4* d = dst + rep * F4_PER_ROW;
      d[lane]      = v0;
      d[lane + 32] = v1;
      d[lane + 64] = v2;
    }
  }
#endif
}

// Zero-fill the pad region t in [3072, 3500) for every batch.
__global__ void __launch_bounds__(256)
length_regulator_pad(float* __restrict__ out) {
  constexpr int PER_B = (T_OUT - T_REP) * F4_PER_ROW;   // 428*96 = 41088 float4
  constexpr int TOTAL = B_ * PER_B;                     // 1,314,816 float4
  const f4v z = {0.f, 0.f, 0.f, 0.f};
  f4v* out4 = reinterpret_cast<f4v*>(out);
  for (int i = blockIdx.x * blockDim.x + threadIdx.x; i < TOTAL;
       i += gridDim.x * blockDim.x) {
    const int b   = i / PER_B;
    const int rem = i - b * PER_B;
    __builtin_nontemporal_store(
        z, out4 + ((size_t)b * T_OUT + T_REP) * F4_PER_ROW + rem);
  }
}

extern "C" void kernel_launch(void* const* d_in, const int* in_sizes, int n_in,
                              void* d_out, int out_size, void* d_ws, size_t ws_size,
                              hipStream_t stream) {
  (void)in_sizes; (void)n_in; (void)out_size; (void)d_ws; (void)ws_size;
  const float* in = (const float*)d_in[0];     // d_in[1] = mel_len scalar (fixed 3500)
  float* out = (float*)d_out;

  // 2048 blocks * 8 waves = 16384 waves -> exactly one input row per wave.
  length_regulator_expand<<<2048, 256, 0, stream>>>(in, out);
  length_regulator_pad<<<1024, 256, 0, stream>>>(out);
}